// GraphConvolution_36000415875227
// MI455X (gfx1250) — compile-verified
//
#include <hip/hip_runtime.h>
#include <hip/hip_bf16.h>

typedef __attribute__((ext_vector_type(2))) float v2f;
typedef __attribute__((ext_vector_type(8))) float v8f;

#define D_IN 256
#define D_OUT 64

// ---------------------------------------------------------------------------
// Kernel 1: support[n, o] = sum_d x[n,d] * W[o,d] + bias[o]
// One wave -> 16 rows x 64 cols using V_WMMA_F32_16X16X4_F32 (full f32).
// Block = 256 threads = 8 waves, each wave a distinct 16-row stripe.
// ---------------------------------------------------------------------------
__global__ void __launch_bounds__(256)
gcn_gemm_wmma(const float* __restrict__ x,
              const float* __restrict__ W,     // [64, 256] row-major
              const float* __restrict__ bias,  // [64]
              float* __restrict__ support,     // [n_nodes, 64]
              int n_nodes)
{
    const int wave = threadIdx.x >> 5;              // 0..7
    const int lane = threadIdx.x & 31;
    const int row0 = (blockIdx.x * 8 + wave) * 16;  // wave-uniform
    if (row0 >= n_nodes) return;                    // EXEC stays all-ones

    // ISA 16x4 f32 A layout: lanes 0-15 hold (M=lane, K=0..1),
    // lanes 16-31 hold (M=lane-16, K=2..3)  -> float2 per lane per k-step.
    const int m    = lane & 15;
    const int ksel = (lane >> 4) << 1;              // 0 or 2

    const float* xrow = x + (size_t)(row0 + m) * D_IN + ksel;
    const float* w0   = W + (size_t)(0 * 16 + m) * D_IN + ksel;
    const float* w1   = W + (size_t)(1 * 16 + m) * D_IN + ksel;
    const float* w2   = W + (size_t)(2 * 16 + m) * D_IN + ksel;
    const float* w3   = W + (size_t)(3 * 16 + m) * D_IN + ksel;

    v8f c0 = {}, c1 = {}, c2 = {}, c3 = {};

    #pragma unroll 4
    for (int k0 = 0; k0 < D_IN; k0 += 4) {
        v2f a  = *(const v2f*)(xrow + k0);
        // B[k][n] = W[n][k]: lanes 0-15 hold (K=0..1, N=lane),
        // lanes 16-31 hold (K=2..3, N=lane-16) -> same float2 addressing.
        v2f b0 = *(const v2f*)(w0 + k0);
        v2f b1 = *(const v2f*)(w1 + k0);
        v2f b2 = *(const v2f*)(w2 + k0);
        v2f b3 = *(const v2f*)(w3 + k0);

        c0 = __builtin_amdgcn_wmma_f32_16x16x4_f32(false, a, false, b0,
                                                   (short)0, c0, false, false);
        c1 = __builtin_amdgcn_wmma_f32_16x16x4_f32(false, a, false, b1,
                                                   (short)0, c1, false, false);
        c2 = __builtin_amdgcn_wmma_f32_16x16x4_f32(false, a, false, b2,
                                                   (short)0, c2, false, false);
        c3 = __builtin_amdgcn_wmma_f32_16x16x4_f32(false, a, false, b3,
                                                   (short)0, c3, false, false);
    }

    // C/D layout: lane<16 -> VGPR r = (M=r, N=lane); lane>=16 -> (M=r+8, N=lane-16)
    const int ncol = lane & 15;
    const int moff = (lane >> 4) * 8;
    v8f acc[4] = {c0, c1, c2, c3};

    #pragma unroll
    for (int t = 0; t < 4; ++t) {
        const int   col = t * 16 + ncol;
        const float bv  = bias[col];
        #pragma unroll
        for (int r = 0; r < 8; ++r) {
            const int row = row0 + moff + r;
            support[(size_t)row * D_OUT + col] = acc[t][r] + bv;
        }
    }
}

// ---------------------------------------------------------------------------
// Kernel 2: out[dst[e]*64 + o] += val[e] * support[src[e]*64 + o]
// 64 consecutive threads per edge: 256B coalesced gather per edge,
// f32 atomic scatter (random dst rows, no structure to exploit).
// ---------------------------------------------------------------------------
__global__ void __launch_bounds__(256)
gcn_scatter(const float* __restrict__ support,
            const int*   __restrict__ esrc,
            const int*   __restrict__ edst,
            const float* __restrict__ eval,
            float* __restrict__ out,
            long long total)   // n_edges * 64
{
    long long gid = (long long)blockIdx.x * blockDim.x + threadIdx.x;
    if (gid >= total) return;
    const int e = (int)(gid >> 6);
    const int o = (int)(gid & 63);

    const int   s = esrc[e];
    const int   d = edst[e];
    const float v = eval[e];

    const float msg = v * support[(size_t)s * D_OUT + o];
    atomicAdd(&out[(size_t)d * D_OUT + o], msg);
}

// ---------------------------------------------------------------------------
// Launcher. Inputs: x, edge_src, edge_dst, edge_val, W_weight, W_bias.
// d_ws holds support [n_nodes, 64] f32 (25.6 MB).
// ---------------------------------------------------------------------------
extern "C" void kernel_launch(void* const* d_in, const int* in_sizes, int n_in,
                              void* d_out, int out_size, void* d_ws, size_t ws_size,
                              hipStream_t stream)
{
    const float* x    = (const float*)d_in[0];
    const int*   esrc = (const int*)  d_in[1];
    const int*   edst = (const int*)  d_in[2];
    const float* evl  = (const float*)d_in[3];
    const float* W    = (const float*)d_in[4];
    const float* bias = (const float*)d_in[5];
    float*       out  = (float*)d_out;
    float*       sup  = (float*)d_ws;

    const int n_nodes = in_sizes[0] / D_IN;   // 100000
    const int n_edges = in_sizes[1];          // 1600000

    // Zero the accumulation target (harness poisons d_out).
    hipMemsetAsync(d_out, 0, (size_t)out_size * sizeof(float), stream);

    // GEMM: 16 rows per wave, 8 waves per block.
    const int row_tiles = (n_nodes + 15) / 16;          // 6250 (exact)
    const int gblocks   = (row_tiles + 7) / 8;
    gcn_gemm_wmma<<<gblocks, 256, 0, stream>>>(x, W, bias, sup, n_nodes);

    // Scatter: one thread per (edge, output channel).
    const long long total  = (long long)n_edges * D_OUT;
    const int       sblocks = (int)((total + 255) / 256);
    gcn_scatter<<<sblocks, 256, 0, stream>>>(sup, esrc, edst, evl, out, total);
}